// CosineAttention_51144470560962
// MI455X (gfx1250) — compile-verified
//
#include <hip/hip_runtime.h>

typedef __attribute__((ext_vector_type(2))) float v2f;
typedef __attribute__((ext_vector_type(8))) float v8f;

#define DIM   384
#define TDIM  1152
#define NTOK  64
#define NHEAD 12
#define HDIM  32
#define NWIMG 256
#define LS    388              // padded LDS row stride (floats): 388 % 64 = 4 -> conflict-free columns
#define SS    68               // padded stride for 64x64 score tile
#define XS_F  (NTOK*LS)        // 24832 floats
#define S_F   (NTOK*SS)        // 4352 floats
#define SMEM_F (3*XS_F + S_F)  // 78848 floats = 315392 bytes (<= 320KB WGP LDS)

__device__ __forceinline__ v8f wmma_f32(v2f a, v2f b, v8f c) {
    // D = A(16x4,f32) * B(4x16,f32) + C(16x16,f32)
    return __builtin_amdgcn_wmma_f32_16x16x4_f32(false, a, false, b, (short)0, c, false, false);
}

__global__ void __launch_bounds__(256, 1)
cosattn_fused(const float* __restrict__ x,
              const float* __restrict__ attn_bias,
              const float* __restrict__ attn_mask,
              const float* __restrict__ W_qkv,
              const float* __restrict__ b_qkv,
              const float* __restrict__ W_proj,
              const float* __restrict__ b_proj,
              const float* __restrict__ tau,
              float* __restrict__ out)
{
    extern __shared__ float smem[];
    float* xs = smem;            // x  -> then q-hat -> then attention output (64 x LS)
    float* ks = smem + XS_F;     // k-hat (64 x LS)
    float* vs = smem + 2*XS_F;   // v     (64 x LS)
    float* sS = smem + 3*XS_F;   // scores / probs (64 x SS)

    const int tid  = threadIdx.x;
    const int lane = tid & 31;
    const int w    = tid >> 5;       // wave 0..7
    const int ln16 = lane & 15;
    const int hi   = lane >> 4;      // half-wave select
    const int k2   = hi << 1;        // K offset (+0 / +2) per WMMA fp32 operand layout
    const int win  = blockIdx.x;

    // ------------------------------------------------------------------
    // Load x window into LDS (vectorized)
    // ------------------------------------------------------------------
    const float* xw = x + (size_t)win * NTOK * DIM;
    for (int i = tid; i < NTOK * (DIM/4); i += 256) {
        const int r  = i / (DIM/4);
        const int c4 = (i - r*(DIM/4)) * 4;
        const float4 v = *(const float4*)(xw + r*DIM + c4);
        xs[r*LS + c4 + 0] = v.x;
        xs[r*LS + c4 + 1] = v.y;
        xs[r*LS + c4 + 2] = v.z;
        xs[r*LS + c4 + 3] = v.w;
    }
    __syncthreads();

    // ------------------------------------------------------------------
    // Phase 1: qkv = x @ W_qkv + b_qkv   (64 x 1152 = 4 M-tiles x 72 N-tiles)
    // q tiles (nt<24) are parked in registers until x is fully consumed,
    // k/v tiles go straight to LDS. Each W element is read exactly once.
    // ------------------------------------------------------------------
    v8f qacc[3][4];
    for (int i = 0; i < 9; ++i) {
        const int nt   = w + i*8;          // 0..71
        const int ncol = nt*16 + ln16;     // column in [0,1152)
        v8f acc[4] = {};
        for (int k = 0; k < DIM; k += 4) {
            const int kr = k + k2;
            v2f b;
            b.x = W_qkv[kr*TDIM + ncol];
            b.y = W_qkv[(kr+1)*TDIM + ncol];
#pragma unroll
            for (int mt = 0; mt < 4; ++mt) {
                const int arow = (mt*16 + ln16)*LS + kr;
                v2f a; a.x = xs[arow]; a.y = xs[arow + 1];
                acc[mt] = wmma_f32(a, b, acc[mt]);
            }
        }
        const float bb = b_qkv[ncol];
        if (i < 3) {                       // q: hold in registers
#pragma unroll
            for (int mt = 0; mt < 4; ++mt)
#pragma unroll
                for (int j = 0; j < 8; ++j)
                    qacc[i][mt][j] = acc[mt][j] + bb;
        } else {                           // k / v: store to LDS
            float* dst    = (i < 6) ? ks : vs;
            const int cc  = ncol - ((i < 6) ? DIM : 2*DIM);
#pragma unroll
            for (int mt = 0; mt < 4; ++mt)
#pragma unroll
                for (int j = 0; j < 8; ++j) {
                    const int row = mt*16 + j + (hi << 3);
                    dst[row*LS + cc] = acc[mt][j] + bb;
                }
        }
    }
    __syncthreads();                       // everyone done reading x
    // flush q over x
    for (int i = 0; i < 3; ++i) {
        const int ncol = (w + i*8)*16 + ln16;
#pragma unroll
        for (int mt = 0; mt < 4; ++mt)
#pragma unroll
            for (int j = 0; j < 8; ++j) {
                const int row = mt*16 + j + (hi << 3);
                xs[row*LS + ncol] = qacc[i][mt][j];
            }
    }
    __syncthreads();

    // ------------------------------------------------------------------
    // Phase 2: per (row, head) L2-normalize q and k; fold 1/tau into q-hat
    // ------------------------------------------------------------------
    for (int t = tid; t < NTOK*NHEAD; t += 256) {
        const int r = t / NHEAD;
        const int h = t - r*NHEAD;
        const int base = r*LS + h*HDIM;
        float ss = 0.f;
#pragma unroll 8
        for (int c = 0; c < HDIM; ++c) { const float qv = xs[base+c]; ss += qv*qv; }
        const float tc = fmaxf(tau[h], 0.001f);
        float sc = 1.f / (fmaxf(sqrtf(ss), 1e-12f) * tc);
#pragma unroll 8
        for (int c = 0; c < HDIM; ++c) xs[base+c] *= sc;
        ss = 0.f;
#pragma unroll 8
        for (int c = 0; c < HDIM; ++c) { const float kv = ks[base+c]; ss += kv*kv; }
        sc = 1.f / fmaxf(sqrtf(ss), 1e-12f);
#pragma unroll 8
        for (int c = 0; c < HDIM; ++c) ks[base+c] *= sc;
    }
    __syncthreads();

    // ------------------------------------------------------------------
    // Phase 3: attention, head by head
    // ------------------------------------------------------------------
    const size_t maskbase = (size_t)(win % NWIMG) * NTOK * NTOK;
    for (int h = 0; h < NHEAD; ++h) {
        const int hb = h * HDIM;
        // S = q_h @ k_h^T  (+ bias + mask): 16 tiles, 2 per wave
        for (int tt = w; tt < 16; tt += 8) {
            const int mt = tt & 3, nt = tt >> 2;
            v8f c = {};
            const int arow = (mt*16 + ln16)*LS + hb + k2;
            const int brow = (nt*16 + ln16)*LS + hb + k2;
#pragma unroll
            for (int kk = 0; kk < HDIM; kk += 4) {
                v2f a; a.x = xs[arow + kk]; a.y = xs[arow + kk + 1];
                v2f b; b.x = ks[brow + kk]; b.y = ks[brow + kk + 1];
                c = wmma_f32(a, b, c);
            }
            const int ncol = nt*16 + ln16;
#pragma unroll
            for (int j = 0; j < 8; ++j) {
                const int row = mt*16 + j + (hi << 3);
                sS[row*SS + ncol] = c[j]
                    + attn_bias[h*NTOK*NTOK + row*NTOK + ncol]
                    + attn_mask[maskbase + row*NTOK + ncol];
            }
        }
        __syncthreads();
        // row softmax (64 rows, one thread each)
        if (tid < NTOK) {
            const int base = tid * SS;
            float m = -3.4e38f;
#pragma unroll 8
            for (int n = 0; n < NTOK; ++n) m = fmaxf(m, sS[base+n]);
            float s = 0.f;
#pragma unroll 8
            for (int n = 0; n < NTOK; ++n) { const float e = __expf(sS[base+n]-m); sS[base+n] = e; s += e; }
            const float inv = 1.f / s;
#pragma unroll 8
            for (int n = 0; n < NTOK; ++n) sS[base+n] *= inv;
        }
        __syncthreads();
        // O_h = P @ v_h : 8 tiles (4x2), one per wave; overwrite q_h region in xs
        {
            const int mt = w & 3, nt = w >> 2;
            v8f c = {};
            const int arow = (mt*16 + ln16)*SS + k2;
            const int bcol = hb + nt*16 + ln16;
#pragma unroll
            for (int kk = 0; kk < NTOK; kk += 4) {
                v2f a; a.x = sS[arow + kk]; a.y = sS[arow + kk + 1];
                const int kr = kk + k2;
                v2f b; b.x = vs[kr*LS + bcol]; b.y = vs[(kr+1)*LS + bcol];
                c = wmma_f32(a, b, c);
            }
#pragma unroll
            for (int j = 0; j < 8; ++j) {
                const int row = mt*16 + j + (hi << 3);
                xs[row*LS + bcol] = c[j];
            }
        }
        __syncthreads();
    }

    // ------------------------------------------------------------------
    // Phase 4: y = O @ W_proj + b_proj  -> global (96 tiles, 12 per wave)
    // ------------------------------------------------------------------
    float* yw = out + (size_t)win * NTOK * DIM;
    for (int i = 0; i < 3; ++i) {
        const int nt   = w + i*8;          // 0..23
        const int ncol = nt*16 + ln16;     // [0,384)
        v8f acc[4] = {};
        for (int k = 0; k < DIM; k += 4) {
            const int kr = k + k2;
            v2f b;
            b.x = W_proj[kr*DIM + ncol];
            b.y = W_proj[(kr+1)*DIM + ncol];
#pragma unroll
            for (int mt = 0; mt < 4; ++mt) {
                const int arow = (mt*16 + ln16)*LS + kr;
                v2f a; a.x = xs[arow]; a.y = xs[arow + 1];
                acc[mt] = wmma_f32(a, b, acc[mt]);
            }
        }
        const float bp = b_proj[ncol];
#pragma unroll
        for (int mt = 0; mt < 4; ++mt)
#pragma unroll
            for (int j = 0; j < 8; ++j) {
                const int row = mt*16 + j + (hi << 3);
                yw[row*DIM + ncol] = acc[mt][j] + bp;
            }
    }
}

extern "C" void kernel_launch(void* const* d_in, const int* in_sizes, int n_in,
                              void* d_out, int out_size, void* d_ws, size_t ws_size,
                              hipStream_t stream) {
    const float* x         = (const float*)d_in[0];
    const float* attn_bias = (const float*)d_in[1];
    const float* attn_mask = (const float*)d_in[2];
    const float* W_qkv     = (const float*)d_in[3];
    const float* b_qkv     = (const float*)d_in[4];
    const float* W_proj    = (const float*)d_in[5];
    const float* b_proj    = (const float*)d_in[6];
    const float* tau       = (const float*)d_in[7];
    float* out             = (float*)d_out;

    const int nwin = in_sizes[0] / (NTOK * DIM);           // 2048 windows
    const size_t smem_bytes = (size_t)SMEM_F * sizeof(float); // 315392 B

    cosattn_fused<<<nwin, 256, smem_bytes, stream>>>(
        x, attn_bias, attn_mask, W_qkv, b_qkv, W_proj, b_proj, tau, out);
}